// ClusteringConsistency_51187420233779
// MI455X (gfx1250) — compile-verified
//
#include <hip/hip_runtime.h>
#include <math.h>

typedef float v2f __attribute__((ext_vector_type(2)));
typedef float v8f __attribute__((ext_vector_type(8)));

#define BSZ   2048
#define DIM   256
#define KPROT 3000
#define QLEN  3840
#define NTOT  5888            // QLEN + BSZ
#define SLD   3008            // padded row stride for S
#define NCHUNK 46             // 5888 / 128 row-chunks for split column reduction
#define NDPART 2048           // sim partials: grid 32x64
#define TEMP_INV 5.0f         // 1/0.2
#define EPS_INV  20.0f        // 1/0.05

// ---------------------------------------------------------------------------
// GEMM1: S[m, k] = exp( (queue_row_m . proto_row_k) / EPS ),  m<3840, k<3000
// 128x64 tile per 128-thread block; 4 waves, each wave = 32(M) x 64(N) strip
// (8 WMMA tiles). V_WMMA_F32_16X16X4_F32 with LDS-staged A/B (stride 33).
// Per k-step/wave: 6 ds-load pairs feed 8 WMMAs.
// ---------------------------------------------------------------------------
__global__ __launch_bounds__(128) void k_gemm_exp(const float* __restrict__ Qm,
                                                  const float* __restrict__ P,
                                                  float* __restrict__ S)
{
    __shared__ float As[128][33];
    __shared__ float Bs[64][33];
    const int m0   = blockIdx.x * 128;
    const int n0   = blockIdx.y * 64;
    const int tid  = threadIdx.x;
    const int lane = tid & 31;
    const int w    = tid >> 5;
    const int l16  = lane & 15;
    const int hi   = lane >> 4;      // 0 or 1

    v8f acc[2][4];
#pragma unroll
    for (int mi = 0; mi < 2; ++mi)
#pragma unroll
        for (int t = 0; t < 4; ++t)
#pragma unroll
            for (int g = 0; g < 8; ++g) acc[mi][t][g] = 0.0f;

    for (int d0 = 0; d0 < DIM; d0 += 32) {
#pragma unroll
        for (int r8 = 0; r8 < 8; ++r8) {
            const int row = (tid >> 3) + r8 * 16;
            const int c4  = (tid & 7) * 4;
            const float4 av = *(const float4*)&Qm[(size_t)(m0 + row) * DIM + d0 + c4];
            As[row][c4+0]=av.x; As[row][c4+1]=av.y; As[row][c4+2]=av.z; As[row][c4+3]=av.w;
        }
#pragma unroll
        for (int r4 = 0; r4 < 4; ++r4) {
            const int row = (tid >> 3) + r4 * 16;
            const int c4  = (tid & 7) * 4;
            float4 bv = make_float4(0.f,0.f,0.f,0.f);
            if (n0 + row < KPROT)
                bv = *(const float4*)&P[(size_t)(n0 + row) * DIM + d0 + c4];
            Bs[row][c4+0]=bv.x; Bs[row][c4+1]=bv.y; Bs[row][c4+2]=bv.z; Bs[row][c4+3]=bv.w;
        }
        __syncthreads();
#pragma unroll
        for (int kk = 0; kk < 32; kk += 4) {
            const int dd = kk + (hi << 1);
            v2f a0, a1;
            a0.x = As[w*32 + l16][dd];      a0.y = As[w*32 + l16][dd + 1];
            a1.x = As[w*32 + 16 + l16][dd]; a1.y = As[w*32 + 16 + l16][dd + 1];
#pragma unroll
            for (int t = 0; t < 4; ++t) {
                v2f b;
                b.x = Bs[t*16 + l16][dd];
                b.y = Bs[t*16 + l16][dd + 1];
                acc[0][t] = __builtin_amdgcn_wmma_f32_16x16x4_f32(
                    false, a0, false, b, (short)0, acc[0][t], false, false);
                acc[1][t] = __builtin_amdgcn_wmma_f32_16x16x4_f32(
                    false, a1, false, b, (short)0, acc[1][t], false, false);
            }
        }
        __syncthreads();
    }

#pragma unroll
    for (int mi = 0; mi < 2; ++mi)
#pragma unroll
        for (int t = 0; t < 4; ++t) {
            const int col = n0 + t*16 + l16;
            if (col < KPROT) {
#pragma unroll
                for (int g = 0; g < 8; ++g) {
                    const int row = m0 + w*32 + mi*16 + g + (hi << 3);
                    S[(size_t)row * SLD + col] = expf(acc[mi][t][g] * EPS_INV);
                }
            }
        }
}

// S rows 3840..5888 come from the crop's output slice.
__global__ __launch_bounds__(256) void k_copy_exp(const float* __restrict__ outs,
                                                  float* __restrict__ S)
{
    const int idx = blockIdx.x * 256 + threadIdx.x;
    if (idx < BSZ * KPROT) {
        const int n = idx / KPROT;
        const int k = idx - n * KPROT;
        S[(size_t)(QLEN + n) * SLD + k] = expf(outs[idx] * EPS_INV);
    }
}

__global__ __launch_bounds__(256) void k_init_c(float* __restrict__ c)
{
    const int i = blockIdx.x * 256 + threadIdx.x;
    if (i < NTOT) c[i] = 1.0f;
}

// Split column reduction: rpart[ch, k] = sum_{n in chunk ch} S[n,k]*c[n].
// grid (12, 46) -> 552 blocks so this 71MB pass can saturate HBM/L2.
__global__ __launch_bounds__(256) void k_rowscale_part(const float* __restrict__ S,
                                                       const float* __restrict__ c,
                                                       float* __restrict__ rpart)
{
    const int k  = blockIdx.x * 256 + threadIdx.x;
    const int ch = blockIdx.y;
    if (k >= KPROT) return;
    float s = 0.f;
    const int nbeg = ch * 128, nend = nbeg + 128;
#pragma unroll 4
    for (int n = nbeg; n < nend; ++n)
        s += S[(size_t)n * SLD + k] * c[n];
    rpart[(size_t)ch * SLD + k] = s;
}

// r[k] = 1 / (K * sum_ch rpart[ch,k])
__global__ __launch_bounds__(256) void k_rowscale_comb(const float* __restrict__ rpart,
                                                       float* __restrict__ r)
{
    const int k = blockIdx.x * 256 + threadIdx.x;
    if (k >= KPROT) return;
    float s = 0.f;
#pragma unroll 2
    for (int ch = 0; ch < NCHUNK; ++ch)
        s += rpart[(size_t)ch * SLD + k];
    r[k] = 1.0f / ((float)KPROT * s);
}

// c[n] = 1 / (NTOT * sum_k S[n,k]*r[k])  (block-per-row, LDS tree reduce)
__global__ __launch_bounds__(256) void k_colscale(const float* __restrict__ S,
                                                  const float* __restrict__ r,
                                                  float* __restrict__ c)
{
    __shared__ float red[256];
    const int n   = blockIdx.x;
    const int tid = threadIdx.x;
    float s = 0.f;
    for (int k = tid; k < KPROT; k += 256)
        s += S[(size_t)n * SLD + k] * r[k];
    red[tid] = s; __syncthreads();
    for (int st = 128; st > 0; st >>= 1) {
        if (tid < st) red[tid] += red[tid + st];
        __syncthreads();
    }
    if (tid == 0) c[n] = 1.0f / ((float)NTOT * red[0]);
}

// Per sample row: q = NTOT*S0*r*c; argmax -> hq, 8 successive minima -> nq
// (argsort ranks 2..7); optional cc term  sum(q*x) - lse(x)*sum(q).
__global__ __launch_bounds__(256) void k_qpost(const float* __restrict__ S,
                                               const float* __restrict__ r,
                                               const float* __restrict__ c,
                                               const float* __restrict__ xrow,
                                               int* __restrict__ hq,
                                               int* __restrict__ nq,
                                               float* __restrict__ cc_part,
                                               int do_cc)
{
    __shared__ float rv[256];
    __shared__ int   ri[256];
    __shared__ int   sel[8];
    const int s   = blockIdx.x;
    const int tid = threadIdx.x;
    const int n   = QLEN + s;
    const float cf = (float)NTOT * c[n];

    float qv[12];
#pragma unroll
    for (int j = 0; j < 12; ++j) {
        const int k = tid + j * 256;
        qv[j] = (k < KPROT) ? cf * S[(size_t)n * SLD + k] * r[k] : 0.0f;
    }

    // ---- argmax (first-index tie-break, like jnp.argmax) ----
    float bv = -1.0f; int bk = KPROT;
#pragma unroll
    for (int j = 0; j < 12; ++j) {
        const int k = tid + j * 256;
        if (k < KPROT && (qv[j] > bv || (qv[j] == bv && k < bk))) { bv = qv[j]; bk = k; }
    }
    rv[tid] = bv; ri[tid] = bk; __syncthreads();
    for (int st = 128; st > 0; st >>= 1) {
        if (tid < st) {
            const float ov = rv[tid + st]; const int ok = ri[tid + st];
            if (ov > rv[tid] || (ov == rv[tid] && ok < ri[tid])) { rv[tid] = ov; ri[tid] = ok; }
        }
        __syncthreads();
    }
    if (tid == 0) hq[s] = ri[0];

    // ---- 8 successive minima (stable ascending order) ----
    for (int p = 0; p < 8; ++p) {
        __syncthreads();
        float mv = 3.4e38f; int mk = KPROT;
#pragma unroll
        for (int j = 0; j < 12; ++j) {
            const int k = tid + j * 256;
            if (k >= KPROT) continue;
            bool used = false;
            for (int u = 0; u < p; ++u) used |= (sel[u] == k);
            if (!used && (qv[j] < mv || (qv[j] == mv && k < mk))) { mv = qv[j]; mk = k; }
        }
        rv[tid] = mv; ri[tid] = mk; __syncthreads();
        for (int st = 128; st > 0; st >>= 1) {
            if (tid < st) {
                const float ov = rv[tid + st]; const int ok = ri[tid + st];
                if (ov < rv[tid] || (ov == rv[tid] && ok < ri[tid])) { rv[tid] = ov; ri[tid] = ok; }
            }
            __syncthreads();
        }
        if (tid == 0) sel[p] = ri[0];
    }
    __syncthreads();
    if (tid < 6) nq[(size_t)s * 6 + tid] = sel[2 + tid];
    __syncthreads();

    // ---- cc term (crop 1 only): sum(q*x) - lse(x)*sum(q), x = output0/TEMP ----
    if (do_cc) {
        float xv[12];
        float lm = -3.4e38f;
#pragma unroll
        for (int j = 0; j < 12; ++j) {
            const int k = tid + j * 256;
            xv[j] = (k < KPROT) ? xrow[(size_t)s * KPROT + k] * TEMP_INV : -3.4e38f;
            lm = fmaxf(lm, xv[j]);
        }
        rv[tid] = lm; __syncthreads();
        for (int st = 128; st > 0; st >>= 1) {
            if (tid < st) rv[tid] = fmaxf(rv[tid], rv[tid + st]);
            __syncthreads();
        }
        const float bm = rv[0]; __syncthreads();

        float se = 0.f, sqx = 0.f, sq = 0.f;
#pragma unroll
        for (int j = 0; j < 12; ++j) {
            const int k = tid + j * 256;
            if (k < KPROT) {
                se  += expf(xv[j] - bm);
                sqx += qv[j] * xv[j];
                sq  += qv[j];
            }
        }
        rv[tid] = se; __syncthreads();
        for (int st = 128; st > 0; st >>= 1) { if (tid < st) rv[tid] += rv[tid + st]; __syncthreads(); }
        const float SE = rv[0]; __syncthreads();
        rv[tid] = sqx; __syncthreads();
        for (int st = 128; st > 0; st >>= 1) { if (tid < st) rv[tid] += rv[tid + st]; __syncthreads(); }
        const float SQX = rv[0]; __syncthreads();
        rv[tid] = sq; __syncthreads();
        for (int st = 128; st > 0; st >>= 1) { if (tid < st) rv[tid] += rv[tid + st]; __syncthreads(); }
        const float SQ = rv[0];
        if (tid == 0) cc_part[s] = SQX - (bm + logf(SE)) * SQ;
    }
}

// normcat[i] = normalized embedding row; i<BSZ -> emb[BSZ+i], else emb[i-BSZ]
__global__ __launch_bounds__(256) void k_norm(const float* __restrict__ emb,
                                              float* __restrict__ ncat)
{
    __shared__ float red[256];
    const int i   = blockIdx.x;
    const int tid = threadIdx.x;
    const int src = (i < BSZ) ? (BSZ + i) : (i - BSZ);
    const float v = emb[(size_t)src * DIM + tid];
    red[tid] = v * v; __syncthreads();
    for (int st = 128; st > 0; st >>= 1) { if (tid < st) red[tid] += red[tid + st]; __syncthreads(); }
    const float inv = rsqrtf(red[0]);
    ncat[(size_t)i * DIM + tid] = v * inv;
}

__global__ __launch_bounds__(256) void k_poslog(const float* __restrict__ ncat,
                                                float* __restrict__ poslog)
{
    __shared__ float red[256];
    const int i   = blockIdx.x;
    const int tid = threadIdx.x;
    red[tid] = ncat[(size_t)i * DIM + tid] * ncat[(size_t)(BSZ + i) * DIM + tid];
    __syncthreads();
    for (int st = 128; st > 0; st >>= 1) { if (tid < st) red[tid] += red[tid + st]; __syncthreads(); }
    if (tid == 0) poslog[i] = red[0] * TEMP_INV;
}

// sim GEMM 4096x4096x256 with fused mask/exp/scalar-reduce epilogue.
// 128x64 tile per block, grid (32, 64).
__global__ __launch_bounds__(128) void k_sim(const float* __restrict__ ncat,
                                             const int* __restrict__ hq,
                                             const int* __restrict__ nq,
                                             float* __restrict__ denom_part)
{
    __shared__ float As[128][33];
    __shared__ float Bs[64][33];
    __shared__ float red[128];
    const int m0   = blockIdx.x * 128;
    const int n0   = blockIdx.y * 64;
    const int tid  = threadIdx.x;
    const int lane = tid & 31;
    const int w    = tid >> 5;
    const int l16  = lane & 15;
    const int hi   = lane >> 4;

    v8f acc[2][4];
#pragma unroll
    for (int mi = 0; mi < 2; ++mi)
#pragma unroll
        for (int t = 0; t < 4; ++t)
#pragma unroll
            for (int g = 0; g < 8; ++g) acc[mi][t][g] = 0.0f;

    for (int d0 = 0; d0 < DIM; d0 += 32) {
#pragma unroll
        for (int r8 = 0; r8 < 8; ++r8) {
            const int row = (tid >> 3) + r8 * 16;
            const int c4  = (tid & 7) * 4;
            const float4 av = *(const float4*)&ncat[(size_t)(m0 + row) * DIM + d0 + c4];
            As[row][c4+0]=av.x; As[row][c4+1]=av.y; As[row][c4+2]=av.z; As[row][c4+3]=av.w;
        }
#pragma unroll
        for (int r4 = 0; r4 < 4; ++r4) {
            const int row = (tid >> 3) + r4 * 16;
            const int c4  = (tid & 7) * 4;
            const float4 bv = *(const float4*)&ncat[(size_t)(n0 + row) * DIM + d0 + c4];
            Bs[row][c4+0]=bv.x; Bs[row][c4+1]=bv.y; Bs[row][c4+2]=bv.z; Bs[row][c4+3]=bv.w;
        }
        __syncthreads();
#pragma unroll
        for (int kk = 0; kk < 32; kk += 4) {
            const int dd = kk + (hi << 1);
            v2f a0, a1;
            a0.x = As[w*32 + l16][dd];      a0.y = As[w*32 + l16][dd + 1];
            a1.x = As[w*32 + 16 + l16][dd]; a1.y = As[w*32 + 16 + l16][dd + 1];
#pragma unroll
            for (int t = 0; t < 4; ++t) {
                v2f b;
                b.x = Bs[t*16 + l16][dd];
                b.y = Bs[t*16 + l16][dd + 1];
                acc[0][t] = __builtin_amdgcn_wmma_f32_16x16x4_f32(
                    false, a0, false, b, (short)0, acc[0][t], false, false);
                acc[1][t] = __builtin_amdgcn_wmma_f32_16x16x4_f32(
                    false, a1, false, b, (short)0, acc[1][t], false, false);
            }
        }
        __syncthreads();
    }

    float local = 0.f;
    int hqj[4];
#pragma unroll
    for (int t = 0; t < 4; ++t) hqj[t] = hq[n0 + t*16 + l16];
#pragma unroll
    for (int mi = 0; mi < 2; ++mi)
#pragma unroll
        for (int g = 0; g < 8; ++g) {
            const int row = m0 + w*32 + mi*16 + g + (hi << 3);
            const int* nr = nq + (size_t)row * 6;
            const int q0 = nr[0], q1 = nr[1], q2 = nr[2], q3 = nr[3], q4 = nr[4], q5 = nr[5];
#pragma unroll
            for (int t = 0; t < 4; ++t) {
                const int col = n0 + t*16 + l16;
                if (col != row) {
                    const int h = hqj[t];
                    if (h == q0 || h == q1 || h == q2 || h == q3 || h == q4 || h == q5)
                        local += expf(acc[mi][t][g] * TEMP_INV);
                }
            }
        }
    red[tid] = local; __syncthreads();
    for (int st = 64; st > 0; st >>= 1) { if (tid < st) red[tid] += red[tid + st]; __syncthreads(); }
    if (tid == 0) denom_part[blockIdx.y * 32 + blockIdx.x] = red[0];
}

// Deterministic final combine: log(denom) - mean(logpos) + 6*cc_loss
__global__ __launch_bounds__(256) void k_final(const float* __restrict__ denom_part,
                                               const float* __restrict__ poslog,
                                               const float* __restrict__ cc_part,
                                               float* __restrict__ out)
{
    __shared__ float r1[256], r2[256], r3[256];
    const int tid = threadIdx.x;
    float a = 0.f, b = 0.f, cc = 0.f;
    for (int i = tid; i < NDPART; i += 256) a  += denom_part[i];
    for (int i = tid; i < BSZ;    i += 256) b  += poslog[i];
    for (int i = tid; i < BSZ;    i += 256) cc += cc_part[i];
    r1[tid] = a; r2[tid] = b; r3[tid] = cc; __syncthreads();
    for (int st = 128; st > 0; st >>= 1) {
        if (tid < st) { r1[tid] += r1[tid+st]; r2[tid] += r2[tid+st]; r3[tid] += r3[tid+st]; }
        __syncthreads();
    }
    if (tid == 0) {
        const float contrast = logf(r1[0]) - r2[0] / (float)BSZ;
        const float cc_loss  = -r3[0] / (float)BSZ;
        out[0] = contrast + 6.0f * cc_loss;
    }
}

extern "C" void kernel_launch(void* const* d_in, const int* in_sizes, int n_in,
                              void* d_out, int out_size, void* d_ws, size_t ws_size,
                              hipStream_t stream)
{
    // inputs: bs, crops_for_assign, nmb_crops_total, queue, prototypes_weight, output, embedding
    const float* queue  = (const float*)d_in[3];   // [2, 3840, 256]
    const float* proto  = (const float*)d_in[4];   // [3000, 256]
    const float* output = (const float*)d_in[5];   // [4096, 3000]
    const float* emb    = (const float*)d_in[6];   // [4096, 256]

    char* ws = (char*)d_ws;
    size_t off = 0;
    auto carve = [&](size_t bytes) -> void* {
        void* p = ws + off;
        off += (bytes + 255) & ~(size_t)255;
        return p;
    };
    float* S          = (float*)carve((size_t)NTOT * SLD * sizeof(float)); // ~70.8 MB
    float* rpart      = (float*)carve((size_t)NCHUNK * SLD * sizeof(float));
    float* r          = (float*)carve((size_t)SLD * sizeof(float));
    float* c          = (float*)carve((size_t)NTOT * sizeof(float));
    int*   hq         = (int*)  carve((size_t)2 * BSZ * sizeof(int));
    int*   nq         = (int*)  carve((size_t)2 * BSZ * 6 * sizeof(int));
    float* cc_part    = (float*)carve((size_t)BSZ * sizeof(float));
    float* ncat       = (float*)carve((size_t)2 * BSZ * DIM * sizeof(float));
    float* poslog     = (float*)carve((size_t)BSZ * sizeof(float));
    float* denom_part = (float*)carve((size_t)NDPART * sizeof(float));
    (void)ws_size;

    for (int crop = 0; crop < 2; ++crop) {
        k_gemm_exp<<<dim3(30, 47), 128, 0, stream>>>(
            queue + (size_t)crop * QLEN * DIM, proto, S);
        k_copy_exp<<<(BSZ * KPROT + 255) / 256, 256, 0, stream>>>(
            output + (size_t)crop * BSZ * KPROT, S);
        k_init_c<<<(NTOT + 255) / 256, 256, 0, stream>>>(c);
        for (int it = 0; it < 3; ++it) {
            k_rowscale_part<<<dim3(12, NCHUNK), 256, 0, stream>>>(S, c, rpart);
            k_rowscale_comb<<<12, 256, 0, stream>>>(rpart, r);
            k_colscale<<<NTOT, 256, 0, stream>>>(S, r, c);
        }
        k_qpost<<<BSZ, 256, 0, stream>>>(
            S, r, c, output /* crop-0 slice is the cc target when crop==1 */,
            hq + (size_t)crop * BSZ, nq + (size_t)crop * BSZ * 6,
            cc_part, (crop == 1) ? 1 : 0);
    }

    k_norm<<<2 * BSZ, 256, 0, stream>>>(emb, ncat);
    k_poslog<<<BSZ, 256, 0, stream>>>(ncat, poslog);
    k_sim<<<dim3(32, 64), 128, 0, stream>>>(ncat, hq, nq, denom_part);
    k_final<<<1, 256, 0, stream>>>(denom_part, poslog, cc_part, (float*)d_out);
}